// WaveletProductLayer_36069135351822
// MI455X (gfx1250) — compile-verified
//
#include <hip/hip_runtime.h>

typedef __attribute__((ext_vector_type(2))) float v2f;
typedef __attribute__((ext_vector_type(4))) float v4f;
typedef __attribute__((ext_vector_type(8))) float v8f;

#define B_DIM 8192
#define N_DIM 512
#define D_DIM 64
#define TB 128        // block tile along B (8 waves x 16)
#define TN 16         // block tile along N (one WMMA tile)
#define XPITCH 132    // 128 + 4 pad: de-conflicts WMMA-A dual-row loads
#define CPITCH 68     // 64 + 4 pad: bank = (4n+d)%64, conflict-free, 16B-aligned rows

__global__ __launch_bounds__(256)
void WaveletProductLayer_kernel(const float* __restrict__ x,
                                const float* __restrict__ centers,
                                const float* __restrict__ scales,
                                float* __restrict__ out) {
  __shared__ __align__(16) float xT[D_DIM][XPITCH]; // x transposed: [d][b_local]
  __shared__ __align__(16) float cq[TN][CPITCH];    // c
  __shared__ __align__(16) float wq[TN][CPITCH];    // 1/s^2
  __shared__ __align__(16) float vq[TN][CPITCH];    // c/s^2
  __shared__ float Karr[TN];                        // sum_d c^2/s^2

  const int t = threadIdx.x;
  const int blockB = blockIdx.x * TB;
  const int blockN = blockIdx.y * TN;

  if (t < TN) Karr[t] = 0.0f;
  __syncthreads();

  // ---- Stage x tile (128 x 64) transposed into LDS; coalesced b128 loads ----
  {
    const int b  = t >> 1;            // 0..127
    const int dh = (t & 1) * 32;      // 0 or 32
    const float* src = x + (size_t)(blockB + b) * D_DIM + dh;
#pragma unroll
    for (int j = 0; j < 8; ++j) {
      v4f val = *(const v4f*)(src + j * 4);
      const int d = dh + j * 4;
      xT[d + 0][b] = val[0];
      xT[d + 1][b] = val[1];
      xT[d + 2][b] = val[2];
      xT[d + 3][b] = val[3];
    }
  }
  // ---- Stage centers/scales (16 x 64): build c, w=1/s^2, v=c*w, K=sum c^2 w ----
  {
    const int n  = t >> 4;            // 0..15
    const int d4 = (t & 15) * 4;      // 0..60
    const float* cs = centers + (size_t)(blockN + n) * D_DIM + d4;
    const float* ss = scales  + (size_t)(blockN + n) * D_DIM + d4;
    v4f c4 = *(const v4f*)cs;
    v4f s4 = *(const v4f*)ss;
    v4f w4, u4;
    float kpart = 0.0f;
#pragma unroll
    for (int j = 0; j < 4; ++j) {
      float inv = 1.0f / (s4[j] * s4[j]);
      w4[j] = inv;
      u4[j] = c4[j] * inv;
      kpart = __builtin_fmaf(c4[j], u4[j], kpart); // c^2 * w
    }
    *(v4f*)&cq[n][d4] = c4;
    *(v4f*)&wq[n][d4] = w4;
    *(v4f*)&vq[n][d4] = u4;
    atomicAdd(&Karr[n], kpart);       // ds_add_f32
  }
  __syncthreads();

  const int wave = t >> 5;            // 0..7 -> which 16-row slab of B
  const int lane = t & 31;
  const int half = lane >> 4;         // k-halves for A/B operands
  const int ln   = lane & 15;         // A row / B column / output n

  // ---- S = (x^2) . w^T,  P = x . v^T  via v_wmma_f32_16x16x4_f32 ----
  v8f S = {};
  v8f P = {};
  const int arow = wave * 16 + ln;
#pragma unroll
  for (int kd = 0; kd < 16; ++kd) {
    const int dA = kd * 4 + half * 2; // lanes 0-15: K=0,1  lanes 16-31: K=2,3
    const float xv0 = xT[dA + 0][arow];
    const float xv1 = xT[dA + 1][arow];
    v2f a2 = {xv0, xv1};
    v2f a1 = {xv0 * xv0, xv1 * xv1};
    v2f b1 = {wq[ln][dA + 0], wq[ln][dA + 1]};
    v2f b2 = {vq[ln][dA + 0], vq[ln][dA + 1]};
    S = __builtin_amdgcn_wmma_f32_16x16x4_f32(false, a1, false, b1,
                                              (short)0, S, false, false);
    P = __builtin_amdgcn_wmma_f32_16x16x4_f32(false, a2, false, b2,
                                              (short)0, P, false, false);
  }

  // ---- Elementwise product  prod_d (1 - z^2)  on the VALU pipe ----
  float prod[8];
#pragma unroll
  for (int i = 0; i < 8; ++i) prod[i] = 1.0f;
  const int pbase = wave * 16 + half * 8; // rows this lane owns (C layout)
#pragma unroll 4
  for (int d = 0; d < D_DIM; ++d) {
    const float cv = cq[ln][d];
    const float wv = wq[ln][d];
    v4f xa = *(const v4f*)&xT[d][pbase];      // broadcast b128 LDS loads
    v4f xb = *(const v4f*)&xT[d][pbase + 4];
#pragma unroll
    for (int i = 0; i < 4; ++i) {
      float da = xa[i] - cv;
      float za = da * da * wv;
      prod[i]     = __builtin_fmaf(-za, prod[i], prod[i]);      // *= (1 - z^2)
      float db = xb[i] - cv;
      float zb = db * db * wv;
      prod[i + 4] = __builtin_fmaf(-zb, prod[i + 4], prod[i + 4]);
    }
  }

  // ---- Combine: h = prod * exp(-0.5 * (S - 2P + K)) ----
  const float Kn = Karr[ln];
  const float nHalfLog2e = -0.72134752044448170f; // -0.5 * log2(e)
  const int outCol = blockN + ln;
#pragma unroll
  for (int i = 0; i < 8; ++i) {
    float sumz2 = S[i] - 2.0f * P[i] + Kn;
    float hval  = prod[i] * exp2f(nHalfLog2e * sumz2);
    const int brow = blockB + wave * 16 + half * 8 + i;
    out[(size_t)brow * N_DIM + outCol] = hval;
  }
}

extern "C" void kernel_launch(void* const* d_in, const int* in_sizes, int n_in,
                              void* d_out, int out_size, void* d_ws, size_t ws_size,
                              hipStream_t stream) {
  const float* x       = (const float*)d_in[0]; // (8192, 64)
  const float* centers = (const float*)d_in[1]; // (512, 64)
  const float* scales  = (const float*)d_in[2]; // (512, 64)
  float* out = (float*)d_out;                   // (8192, 512)
  (void)in_sizes; (void)n_in; (void)out_size; (void)d_ws; (void)ws_size;

  dim3 grid(B_DIM / TB, N_DIM / TN); // 64 x 32
  dim3 block(256);                   // 8 wave32s
  WaveletProductLayer_kernel<<<grid, block, 0, stream>>>(x, centers, scales, out);
}